// DetNMSPostProcessor_60473139528492
// MI455X (gfx1250) — compile-verified
//
#include <hip/hip_runtime.h>
#include <stdint.h>
#include <stddef.h>

// ---------------------------------------------------------------------------
// DETR-style NMS post-processing for MI455X (gfx1250, wave32).
//   Phase 1 (prep_kernel): bandwidth-bound streaming pass over logits/boxes.
//   Phase 2 (nms_kernel):  one workgroup per image, everything in LDS:
//       bitonic sort -> async global->LDS gather of sorted boxes
//       -> greedy suppression -> wave32 ballot compaction.
// CDNA5-specific paths used: global_load_async_to_lds_b128 (ASYNCcnt),
// s_wait_asynccnt, wave32 ballot, ~110KB LDS working set.
// ---------------------------------------------------------------------------

#define IOU_THR   0.7f
#define SCORE_THR 0.01f
#define KEEP_TOPK 300
#define BATCH     16
#define NBOX      2048
#define NCLS      80
#define NTHREADS  1024

// Monotone float -> uint32 key (ascending order preserved).
__device__ __forceinline__ unsigned f2key(float f) {
  unsigned u = __float_as_uint(f);
  return u ^ ((u >> 31) ? 0xFFFFFFFFu : 0x80000000u);
}
__device__ __forceinline__ float key2f(unsigned k) {
  unsigned u = (k & 0x80000000u) ? (k ^ 0x80000000u) : ~k;
  return __uint_as_float(u);
}

// ---------------------------------------------------------------------------
// Phase 1: per-box class max/argmax + sigmoid + box convert/scale.
// Pure streaming: 10.5MB logits read with float4 loads, ~0.5us at 23.3 TB/s.
// ---------------------------------------------------------------------------
__global__ __launch_bounds__(256) void prep_kernel(
    const float* __restrict__ logits, const float* __restrict__ boxes,
    const float* __restrict__ sizes, float* __restrict__ o_scores,
    int* __restrict__ o_labels, float4* __restrict__ o_boxes) {
  int gid = blockIdx.x * blockDim.x + threadIdx.x;
  if (gid >= BATCH * NBOX) return;
  int b = gid / NBOX;

  const float4* lg = reinterpret_cast<const float4*>(logits + (size_t)gid * NCLS);
  float best = -INFINITY;
  int lbl = 0;
#pragma unroll
  for (int c = 0; c < NCLS / 4; ++c) {
    float4 v = lg[c];
    if (v.x > best) { best = v.x; lbl = 4 * c + 0; }   // strict > : first argmax
    if (v.y > best) { best = v.y; lbl = 4 * c + 1; }
    if (v.z > best) { best = v.z; lbl = 4 * c + 2; }
    if (v.w > best) { best = v.w; lbl = 4 * c + 3; }
  }
  float s = 1.0f / (1.0f + __expf(-best));           // sigmoid
  o_scores[gid] = (s > SCORE_THR) ? s : -1.0f;       // thresholded (ref: where)
  o_labels[gid] = lbl;

  float4 bx = reinterpret_cast<const float4*>(boxes)[gid];
  float sw = sizes[2 * b + 0], sh = sizes[2 * b + 1];
  float4 r;
  r.x = (bx.x - 0.5f * bx.z) * sw;
  r.y = (bx.y - 0.5f * bx.w) * sh;
  r.z = (bx.x + 0.5f * bx.z) * sw;
  r.w = (bx.y + 0.5f * bx.w) * sh;
  o_boxes[gid] = r;
}

// ---------------------------------------------------------------------------
// Phase 2: per-image NMS, one 1024-thread workgroup (32 waves) per image.
// LDS working set ~110KB (fits the 320KB/WGP budget of CDNA5).
// ---------------------------------------------------------------------------
__global__ __launch_bounds__(NTHREADS) void nms_kernel(
    const float* __restrict__ scores, const int* __restrict__ labels,
    const float4* __restrict__ boxes, int* __restrict__ out_l,
    float* __restrict__ out_b, float* __restrict__ out_s) {
  __shared__ unsigned long long keys[NBOX];   // 16 KB sort keys
  __shared__ float4 sbox[NBOX];               // 32 KB sorted boxes (no offset)
  __shared__ float4 obox[NBOX];               // 32 KB class-offset boxes
  __shared__ float  sarea[NBOX];              //  8 KB areas
  __shared__ float  sscore[NBOX];             //  8 KB sorted thresholded scores
  __shared__ int    slab[NBOX];               //  8 KB sorted labels
  __shared__ unsigned char keepf[NBOX];       //  2 KB keep flags (byte-owned)
  __shared__ float  red[NTHREADS];            //  4 KB max-reduction scratch

  const int b = blockIdx.x;
  const int t = threadIdx.x;
  const int base = b * NBOX;

  // ---- 1. build 64-bit sort keys: (score key desc, tie -> lower index first)
  for (int e = t; e < NBOX; e += NTHREADS) {
    float s = scores[base + e];
    keys[e] = ((unsigned long long)f2key(s) << 32) |
              (unsigned)(0xFFFFFFFFu - (unsigned)e);
  }
  __syncthreads();

  // ---- 2. bitonic sort, descending, 2048 elems / 1024 threads (66 stages)
  for (unsigned k = 2; k <= NBOX; k <<= 1) {
    for (unsigned j = k >> 1; j > 0; j >>= 1) {
      unsigned i = (((unsigned)t & ~(j - 1)) << 1) | ((unsigned)t & (j - 1));
      unsigned p = i | j;
      unsigned long long a = keys[i], c = keys[p];
      bool desc = ((i & k) == 0);
      if (desc ? (a < c) : (a > c)) { keys[i] = c; keys[p] = a; }
      __syncthreads();
    }
  }

  // ---- 3. gather sorted labels (VMEM) and boxes (async global->LDS, b128)
  const float4* boxes_b = boxes + base;
  for (int e = t; e < NBOX; e += NTHREADS) {
    unsigned long long kk = keys[e];
    int idx = (int)(0xFFFFFFFFu - (unsigned)(kk & 0xFFFFFFFFu));
    sscore[e] = key2f((unsigned)(kk >> 32));
    slab[e] = labels[base + idx];
    // per-lane gather of 16B box records straight into LDS (ASYNCcnt path)
    unsigned ldsoff = (unsigned)(size_t)(&sbox[e]);   // low 32 bits = LDS offset
    unsigned voff = (unsigned)idx * 16u;              // byte offset from saddr
    asm volatile("global_load_async_to_lds_b128 %0, %1, %2"
                 :: "v"(ldsoff), "v"(voff), "s"(boxes_b)
                 : "memory");
  }
  asm volatile("s_wait_asynccnt 0" ::: "memory");
  __syncthreads();

  // ---- 4. max over all box coords (for the per-class batched-NMS offset)
  float m = -INFINITY;
  for (int e = t; e < NBOX; e += NTHREADS) {
    float4 v = sbox[e];
    m = fmaxf(m, fmaxf(fmaxf(v.x, v.y), fmaxf(v.z, v.w)));
  }
  red[t] = m;
  __syncthreads();
  for (int s = NTHREADS / 2; s > 0; s >>= 1) {
    if (t < s) red[t] = fmaxf(red[t], red[t + s]);
    __syncthreads();
  }
  const float offscale = red[0] + 1.0f;

  // ---- 5. offset boxes, areas, initial keep flags
  for (int e = t; e < NBOX; e += NTHREADS) {
    float off = (float)slab[e] * offscale;
    float4 v = sbox[e];
    float4 o = make_float4(v.x + off, v.y + off, v.z + off, v.w + off);
    obox[e] = o;
    sarea[e] = (o.z - o.x) * (o.w - o.y);
    keepf[e] = (sscore[e] > SCORE_THR) ? (unsigned char)1 : (unsigned char)0;
  }
  __syncthreads();

  // ---- 6. greedy suppression (sequential pivots, parallel victims)
  for (int i = 0; i < NBOX - 1; ++i) {
    if (keepf[i]) {                    // LDS broadcast read, uniform value
      float4 bi = obox[i];
      float  ai = sarea[i];
      for (int j = i + 1 + t; j < NBOX; j += NTHREADS) {
        if (!keepf[j]) continue;       // each thread owns its own bytes j
        float4 bj = obox[j];
        float lx = fmaxf(bi.x, bj.x), ly = fmaxf(bi.y, bj.y);
        float rx = fminf(bi.z, bj.z), ry = fminf(bi.w, bj.w);
        float iw = fmaxf(rx - lx, 0.0f), ih = fmaxf(ry - ly, 0.0f);
        float inter = iw * ih;
        float uni = ai + sarea[j] - inter;
        float iou = (uni > 0.0f) ? (inter / uni) : 0.0f;
        if (iou > IOU_THR) keepf[j] = 0;
      }
    }
    __syncthreads();
  }

  // ---- 7. pad all K output slots first
  for (int r = t; r < KEEP_TOPK; r += NTHREADS) {
    out_l[b * KEEP_TOPK + r] = -1;
    out_s[b * KEEP_TOPK + r] = 0.0f;
    float* ob = out_b + (size_t)(b * KEEP_TOPK + r) * 4;
    ob[0] = 0.0f; ob[1] = 0.0f; ob[2] = 0.0f; ob[3] = 0.0f;
  }
  __syncthreads();

  // ---- 8. wave32 ballot compaction: first KEEP_TOPK survivors in order
  if (t < 32) {
    int cnt = 0;
    for (int c = 0; c < NBOX / 32 && cnt < KEEP_TOPK; ++c) {
      int e = c * 32 + t;
      bool kp = keepf[e] != 0;
      unsigned mask = (unsigned)__ballot(kp);            // wave32: low 32 bits
      int rank = cnt + __popc(mask & ((1u << t) - 1u));
      if (kp && rank < KEEP_TOPK) {
        out_l[b * KEEP_TOPK + rank] = slab[e];
        out_s[b * KEEP_TOPK + rank] = sscore[e];
        float4 v = sbox[e];
        float* ob = out_b + (size_t)(b * KEEP_TOPK + rank) * 4;
        ob[0] = v.x; ob[1] = v.y; ob[2] = v.z; ob[3] = v.w;
      }
      cnt += __popc(mask);
    }
  }
}

// ---------------------------------------------------------------------------
extern "C" void kernel_launch(void* const* d_in, const int* in_sizes, int n_in,
                              void* d_out, int out_size, void* d_ws, size_t ws_size,
                              hipStream_t stream) {
  (void)in_sizes; (void)n_in; (void)out_size; (void)ws_size;

  const float* logits = (const float*)d_in[0];   // [B,N,C] f32
  const float* boxes  = (const float*)d_in[1];   // [B,N,4] f32 (cxcywh)
  const float* sizes  = (const float*)d_in[2];   // [B,2]   f32

  const int BN = BATCH * NBOX;
  float*  ws_scores = (float*)d_ws;                               // BN f32
  int*    ws_labels = (int*)((char*)d_ws + (size_t)BN * 4);       // BN i32
  float4* ws_boxes  = (float4*)((char*)d_ws + (size_t)BN * 8);    // BN f32x4

  // Output layout (reference return order, concatenated flat):
  //   labels [B,300] i32 | boxes [B,300,4] f32 | scores [B,300] f32
  int*   out_l = (int*)d_out;
  float* out_b = (float*)d_out + BATCH * KEEP_TOPK;
  float* out_s = (float*)d_out + BATCH * KEEP_TOPK * 5;

  prep_kernel<<<BN / 256, 256, 0, stream>>>(logits, boxes, sizes,
                                            ws_scores, ws_labels, ws_boxes);
  nms_kernel<<<BATCH, NTHREADS, 0, stream>>>(ws_scores, ws_labels, ws_boxes,
                                             out_l, out_b, out_s);
}